// AcceleratedAttentionPool1d_48850958025247
// MI455X (gfx1250) — compile-verified
//
#include <hip/hip_runtime.h>

typedef __attribute__((ext_vector_type(2))) float v2f;
typedef __attribute__((ext_vector_type(8))) float v8f;

#define B_DIM 8
#define E_DIM 256
#define S_DIM 2048
#define KW    9
#define PADW  4
#define INV_SCALE (1.0f / 24.0f)   // 1/(sqrt(256)*1.5)

// ---------------------------------------------------------------------------
// Kernel 1: Y[b,t,f] = sum_e x[b,e,t] * Wq[f,e] + bq[f]
// One block = one 16-wide t-tile of one batch.
// The 16x256 x-tile (A operand, shared by all 16 f-tiles) is staged ONCE into
// LDS via async global->LDS B128 copies (ASYNCcnt), then 8 waves x 2 f-tiles
// run the K-loop of fp32 WMMA 16x16x4 with A from LDS, B (Wq rows) from L2.
// ---------------------------------------------------------------------------
__global__ __launch_bounds__(256) void qproj_kernel(
    const float* __restrict__ x, const float* __restrict__ Wq,
    const float* __restrict__ bq, float* __restrict__ Y)
{
    __shared__ float xs[16 * E_DIM];              // [e][t] rows of 16 floats (16 KB)

    const int tile = blockIdx.x;                  // 0 .. B*S/16-1
    const int b    = tile / (S_DIM / 16);
    const int t0   = (tile % (S_DIM / 16)) * 16;
    const int tid  = threadIdx.x;
    const float* xb = x + (size_t)b * E_DIM * S_DIM + t0;

    // ---- async stage: 1024 x 16B chunks, 4 per thread, global -> LDS ----
    const uint32_t lds0 = (uint32_t)(uintptr_t)&xs[0];   // LDS byte address
    #pragma unroll
    for (int i = 0; i < 4; ++i) {
        const int c    = tid + 256 * i;           // chunk id
        const int e    = c >> 2;                  // feature row
        const int part = c & 3;                   // 16B chunk within 64B row
        const float*   g = xb + (size_t)e * S_DIM + part * 4;
        const uint32_t l = lds0 + (uint32_t)(e * 16 + part * 4) * 4u;
        asm volatile("global_load_async_to_lds_b128 %0, %1, off"
                     :: "v"(l), "v"(g) : "memory");
    }
    asm volatile("s_wait_asynccnt 0x0" ::: "memory");
    __syncthreads();                              // all waves' chunks visible

    const int wave = tid >> 5;
    const int lane = tid & 31;
    const int mn   = lane & 15;                   // A-row m / B-col n / D-col n
    const int kb   = (lane >> 4) * 2;             // K sub-offset per half-wave
    const int half = lane >> 4;

    #pragma unroll
    for (int ft = 0; ft < 2; ++ft) {
        const int f0 = (wave * 2 + ft) * 16;
        const float* wrow = Wq + (size_t)(f0 + mn) * E_DIM;   // B[k,n] = Wq[f0+n, e]
        v8f acc = {};
        #pragma unroll 4
        for (int k0 = 0; k0 < E_DIM; k0 += 4) {
            v2f a;                                // A[m,k] = xs[k][m], conflict-free
            a.x = xs[(k0 + kb) * 16 + mn];
            a.y = xs[(k0 + kb + 1) * 16 + mn];
            v2f bv = *(const v2f*)(wrow + k0 + kb);
            acc = __builtin_amdgcn_wmma_f32_16x16x4_f32(
                      false, a, false, bv, (short)0, acc, false, false);
        }
        const float bias = bq[f0 + mn];
        float* yb = Y + ((size_t)b * S_DIM + t0) * E_DIM + f0;
        #pragma unroll
        for (int j = 0; j < 8; ++j) {
            const int mm = j + half * 8;          // D row = t offset
            yb[(size_t)mm * E_DIM + mn] = acc[j] + bias;
        }
    }
}

// ---------------------------------------------------------------------------
// Kernel 2 (fused): window energies -> softmax -> context (LDS) -> Wo GEMM.
// One block = one 16-wide t-tile of one batch.
//  Phase A: 16 groups of 16 lanes; group g owns s = t0+g.
//  Phase B: 8 waves x 2 f-tiles of out[b,f,t] = (Wo . ctx + bo)/9 via WMMA.
// ---------------------------------------------------------------------------
__global__ __launch_bounds__(256) void attn_out_kernel(
    const float* __restrict__ x, const float* __restrict__ Y,
    const float* __restrict__ bq, const float* __restrict__ Wo,
    const float* __restrict__ bo, float* __restrict__ out)
{
    __shared__ float ctx[16][E_DIM + 4];          // +4 pad: bank stride 4, no conflicts

    const int tile = blockIdx.x;
    const int b    = tile / (S_DIM / 16);
    const int t0   = (tile % (S_DIM / 16)) * 16;
    const int tid  = threadIdx.x;
    const int g      = tid >> 4;                  // group -> s index in tile
    const int lane16 = tid & 15;
    const int s      = t0 + g;

    const float* yb = Y + (size_t)b * S_DIM * E_DIM;
    const float* ys = yb + (size_t)s * E_DIM;
    const float* xb = x + (size_t)b * E_DIM * S_DIM;

    // --- energies: en[l] = <y[s], y[s+l-4]> / 24  (OOB row of Q == bq) ---
    float en[KW];
    #pragma unroll
    for (int l = 0; l < KW; ++l) {
        const int sl = s + l - PADW;
        const float* yl = (sl >= 0 && sl < S_DIM) ? (yb + (size_t)sl * E_DIM) : bq;
        float p = 0.f;
        #pragma unroll 4
        for (int e = lane16; e < E_DIM; e += 16)
            p += ys[e] * yl[e];
        #pragma unroll
        for (int off = 8; off >= 1; off >>= 1)    // xor tree: all 16 lanes get sum
            p += __shfl_xor(p, off, 16);
        en[l] = p * INV_SCALE;
    }

    // --- softmax over the 9 window slots (redundant in all 16 lanes) ---
    float mx = en[0];
    #pragma unroll
    for (int l = 1; l < KW; ++l) mx = fmaxf(mx, en[l]);
    float w[KW], sum = 0.f;
    #pragma unroll
    for (int l = 0; l < KW; ++l) { w[l] = __expf(en[l] - mx); sum += w[l]; }
    const float inv = 1.0f / sum;

    // --- context: ctx[s,e] = sum_l w[l] * x[b,e,s+l-4]  (zero-padded) ---
    #pragma unroll 4
    for (int e = lane16; e < E_DIM; e += 16) {
        const float* xrow = xb + (size_t)e * S_DIM;
        float c = 0.f;
        #pragma unroll
        for (int l = 0; l < KW; ++l) {
            const int sl = s + l - PADW;
            if (sl >= 0 && sl < S_DIM) c += w[l] * inv * xrow[sl];
        }
        ctx[g][e] = c;
    }
    __syncthreads();

    // --- output GEMM: out[b, f0+m, t0+n] = (sum_e Wo[f,e]*ctx[n][e] + bo[f])/9
    const int wave = tid >> 5;
    const int lane = tid & 31;
    const int mn   = lane & 15;
    const int kb   = (lane >> 4) * 2;
    const int half = lane >> 4;

    #pragma unroll
    for (int ft = 0; ft < 2; ++ft) {
        const int f0 = (wave * 2 + ft) * 16;
        const float* arow = Wo + (size_t)(f0 + mn) * E_DIM;   // A[m,k] = Wo[f0+m, e]
        v8f acc = {};
        #pragma unroll 4
        for (int k0 = 0; k0 < E_DIM; k0 += 4) {
            v2f a  = *(const v2f*)(arow + k0 + kb);
            v2f bv = *(const v2f*)&ctx[mn][k0 + kb];          // B[k,n] = ctx[n][e]
            acc = __builtin_amdgcn_wmma_f32_16x16x4_f32(
                      false, a, false, bv, (short)0, acc, false, false);
        }
        float* ob = out + ((size_t)b * E_DIM + f0) * S_DIM + t0;
        #pragma unroll
        for (int j = 0; j < 8; ++j) {
            const int mm = j + half * 8;          // D row = f offset
            ob[(size_t)mm * S_DIM + mn] = (acc[j] + bo[f0 + mm]) * (1.0f / KW);
        }
    }
}

// ---------------------------------------------------------------------------
extern "C" void kernel_launch(void* const* d_in, const int* in_sizes, int n_in,
                              void* d_out, int out_size, void* d_ws, size_t ws_size,
                              hipStream_t stream)
{
    const float* x  = (const float*)d_in[0];
    const float* Wq = (const float*)d_in[1];
    const float* bq = (const float*)d_in[2];
    const float* Wo = (const float*)d_in[3];
    const float* bo = (const float*)d_in[4];
    float* out = (float*)d_out;
    float* Y   = (float*)d_ws;     // B*S*E floats = 16 MB scratch

    const int nblk = B_DIM * (S_DIM / 16);   // 1024 blocks of 256 threads
    qproj_kernel<<<dim3(nblk), dim3(256), 0, stream>>>(x, Wq, bq, Y);
    attn_out_kernel<<<dim3(nblk), dim3(256), 0, stream>>>(x, Y, bq, Wo, bo, out);
}